// HgnnEncoder_15642270892331
// MI455X (gfx1250) — compile-verified
//
#include <hip/hip_runtime.h>

#define N_NODES   100000
#define N_HE      30000
#define N_INC     300000
#define IN_CH     128
#define HID       256
#define OUT_CH    128
#define EPS_BN    1e-5f
#define BN_CHUNK  125   // 100000 / 125 = 800 blocks

typedef float v2f __attribute__((ext_vector_type(2)));
typedef float v8f __attribute__((ext_vector_type(8)));

// ---------------------------------------------------------------- utilities
__global__ __launch_bounds__(256) void zero_kernel(float* __restrict__ p, int n) {
    int i = blockIdx.x * 256 + threadIdx.x;
    int stride = gridDim.x * 256;
    for (; i < n; i += stride) p[i] = 0.0f;
}

__global__ __launch_bounds__(256) void degree_kernel(const int* __restrict__ node_idx,
                                                     const int* __restrict__ he_idx,
                                                     float* __restrict__ dn,
                                                     float* __restrict__ de) {
    int i = blockIdx.x * 256 + threadIdx.x;
    if (i >= N_INC) return;
    unsafeAtomicAdd(&dn[node_idx[i]], 1.0f);
    unsafeAtomicAdd(&de[he_idx[i]], 1.0f);
}

__global__ __launch_bounds__(256) void invert_kernel(float* __restrict__ p, int n) {
    int i = blockIdx.x * 256 + threadIdx.x;
    if (i >= n) return;
    float v = p[i];
    p[i] = (v > 0.0f) ? (1.0f / v) : 0.0f;
}

// ---------------------------------------------------------------- WMMA GEMM
// One wave computes a 16(M) x 64(N) tile of C = A[NxK] * B[KxF] (row-major fp32)
// using 4 accumulators; the A fragment is shared by 4 back-to-back WMMAs.
// V_WMMA_F32_16X16X4_F32 layouts (wave32):
//   A 16x4 : lanes 0-15 -> row=lane, K=0,1 in v0,v1 ; lanes 16-31 -> K=2,3
//   B 4x16 : lanes 0-15 -> col=lane, K=0,1 in v0,v1 ; lanes 16-31 -> K=2,3
//   C 16x16: vgpr v -> row = v + 8*(lane>=16), col = lane&15
__global__ __launch_bounds__(32) void gemm_wmma(const float* __restrict__ A,
                                                const float* __restrict__ B,
                                                float* __restrict__ C,
                                                int K, int F) {
    const int lane = threadIdx.x;
    const int hw   = lane >> 4;      // half-wave: 0 or 1
    const int l    = lane & 15;
    const int m0   = blockIdx.x * 16;
    const int n0   = blockIdx.y * 64;

    const float* arow  = A + (size_t)(m0 + l) * K + (hw << 1);
    const float* bbase = B + (size_t)(hw << 1) * F + (n0 + l);

    v8f acc0 = {0.f, 0.f, 0.f, 0.f, 0.f, 0.f, 0.f, 0.f};
    v8f acc1 = acc0, acc2 = acc0, acc3 = acc0;

    for (int k = 0; k < K; k += 4) {
        v2f a;
        a.x = arow[k];
        a.y = arow[k + 1];
        const float* bk = bbase + (size_t)k * F;
        v2f b0, b1, b2, b3;
        b0.x = bk[0];       b0.y = bk[(size_t)F + 0];
        b1.x = bk[16];      b1.y = bk[(size_t)F + 16];
        b2.x = bk[32];      b2.y = bk[(size_t)F + 32];
        b3.x = bk[48];      b3.y = bk[(size_t)F + 48];
        acc0 = __builtin_amdgcn_wmma_f32_16x16x4_f32(false, a, false, b0, (short)0, acc0, false, false);
        acc1 = __builtin_amdgcn_wmma_f32_16x16x4_f32(false, a, false, b1, (short)0, acc1, false, false);
        acc2 = __builtin_amdgcn_wmma_f32_16x16x4_f32(false, a, false, b2, (short)0, acc2, false, false);
        acc3 = __builtin_amdgcn_wmma_f32_16x16x4_f32(false, a, false, b3, (short)0, acc3, false, false);
    }

    float* crow = C + (size_t)(m0 + hw * 8) * F + (n0 + l);
#pragma unroll
    for (int v = 0; v < 8; ++v) {
        crow[(size_t)v * F +  0] = acc0[v];
        crow[(size_t)v * F + 16] = acc1[v];
        crow[(size_t)v * F + 32] = acc2[v];
        crow[(size_t)v * F + 48] = acc3[v];
    }
}

// ---------------------------------------------------------------- scatters
// node -> hyperedge: he[e] += xw[n]   (float4 per thread, 4 fp32 atomics)
__global__ __launch_bounds__(256) void scatter_fwd(const float* __restrict__ src,
                                                   const int* __restrict__ node_idx,
                                                   const int* __restrict__ he_idx,
                                                   float* __restrict__ dst,
                                                   int F, int gshift) {
    int idx = blockIdx.x * 256 + threadIdx.x;
    int total = N_INC << gshift;
    if (idx >= total) return;
    int i = idx >> gshift;
    int g = (idx & ((1 << gshift) - 1)) << 2;
    int n = node_idx[i];
    int e = he_idx[i];
    const float4 v = *(const float4*)(src + (size_t)n * F + g);
    float* d = dst + (size_t)e * F + g;
    unsafeAtomicAdd(d + 0, v.x);
    unsafeAtomicAdd(d + 1, v.y);
    unsafeAtomicAdd(d + 2, v.z);
    unsafeAtomicAdd(d + 3, v.w);
}

// hyperedge -> node, with B^{-1} fused into the gather: agg[n] += he[e] * binv[e]
__global__ __launch_bounds__(256) void scatter_bwd(const float* __restrict__ he,
                                                   const float* __restrict__ binv,
                                                   const int* __restrict__ node_idx,
                                                   const int* __restrict__ he_idx,
                                                   float* __restrict__ dst,
                                                   int F, int gshift) {
    int idx = blockIdx.x * 256 + threadIdx.x;
    int total = N_INC << gshift;
    if (idx >= total) return;
    int i = idx >> gshift;
    int g = (idx & ((1 << gshift) - 1)) << 2;
    int n = node_idx[i];
    int e = he_idx[i];
    float s = binv[e];
    const float4 v = *(const float4*)(he + (size_t)e * F + g);
    float* d = dst + (size_t)n * F + g;
    unsafeAtomicAdd(d + 0, v.x * s);
    unsafeAtomicAdd(d + 1, v.y * s);
    unsafeAtomicAdd(d + 2, v.z * s);
    unsafeAtomicAdd(d + 3, v.w * s);
}

// ---------------------------------------------------------------- epilogues
// h = relu(agg * Dinv + b) in place, accumulate per-channel sum / sumsq (F == 256)
__global__ __launch_bounds__(256) void epilogue_bn(float* __restrict__ h,
                                                   const float* __restrict__ dinv,
                                                   const float* __restrict__ bias,
                                                   float* __restrict__ csum,
                                                   float* __restrict__ csumsq) {
    const int c = threadIdx.x;
    const float b = bias[c];
    int n = blockIdx.x * BN_CHUNK;
    int nend = n + BN_CHUNK;
    if (nend > N_NODES) nend = N_NODES;
    float s = 0.0f, s2 = 0.0f;
    for (; n < nend; ++n) {
        float v = fmaxf(h[(size_t)n * HID + c] * dinv[n] + b, 0.0f);
        h[(size_t)n * HID + c] = v;
        s += v;
        s2 += v * v;
    }
    unsafeAtomicAdd(&csum[c], s);
    unsafeAtomicAdd(&csumsq[c], s2);
}

__global__ void bn_stats(const float* __restrict__ csum, const float* __restrict__ csumsq,
                         const float* __restrict__ g, const float* __restrict__ be,
                         float* __restrict__ scale, float* __restrict__ shift) {
    int c = threadIdx.x;
    const float invN = 1.0f / (float)N_NODES;
    float m   = csum[c] * invN;
    float var = csumsq[c] * invN - m * m;
    float inv = rsqrtf(var + EPS_BN);
    float sc  = inv * g[c];
    scale[c] = sc;
    shift[c] = be[c] - m * sc;
}

// one block (256 threads) per node row, F == 256 so c == threadIdx.x
__global__ __launch_bounds__(256) void bn_apply(float* __restrict__ h,
                                                const float* __restrict__ scale,
                                                const float* __restrict__ shift) {
    size_t idx = (size_t)blockIdx.x * 256 + threadIdx.x;
    int c = threadIdx.x;
    h[idx] = h[idx] * scale[c] + shift[c];
}

// layer 3 epilogue: out = relu(agg * Dinv + b3), F == 128, one block per node
__global__ __launch_bounds__(128) void epilogue_out(const float* __restrict__ agg,
                                                    const float* __restrict__ dinv,
                                                    const float* __restrict__ bias,
                                                    float* __restrict__ out) {
    int n = blockIdx.x;
    int c = threadIdx.x;
    float v = agg[(size_t)n * OUT_CH + c] * dinv[n] + bias[c];
    out[(size_t)n * OUT_CH + c] = fmaxf(v, 0.0f);
}

// ---------------------------------------------------------------- driver
extern "C" void kernel_launch(void* const* d_in, const int* in_sizes, int n_in,
                              void* d_out, int out_size, void* d_ws, size_t ws_size,
                              hipStream_t stream) {
    (void)in_sizes; (void)n_in; (void)out_size; (void)ws_size;

    const float* x        = (const float*)d_in[0];
    const int*   edge     = (const int*)d_in[1];
    const int*   node_idx = edge;
    const int*   he_idx   = edge + N_INC;
    const float* W1  = (const float*)d_in[2];
    const float* b1  = (const float*)d_in[3];
    const float* g1  = (const float*)d_in[4];
    const float* be1 = (const float*)d_in[5];
    const float* W2  = (const float*)d_in[6];
    const float* b2  = (const float*)d_in[7];
    const float* g2  = (const float*)d_in[8];
    const float* be2 = (const float*)d_in[9];
    const float* W3  = (const float*)d_in[10];
    const float* b3  = (const float*)d_in[11];
    float* out = (float*)d_out;

    // workspace carve-up
    float* xw     = (float*)d_ws;                   // [N_NODES, 256]
    float* heb    = xw  + (size_t)N_NODES * HID;    // [N_HE, 256]
    float* h      = heb + (size_t)N_HE * HID;       // [N_NODES, 256] (agg / layer output)
    float* dinv   = h   + (size_t)N_NODES * HID;    // [N_NODES]
    float* binv   = dinv + N_NODES;                 // [N_HE]
    float* csum   = binv + N_HE;                    // [256]
    float* csumsq = csum + HID;                     // [256]
    float* scale  = csumsq + HID;                   // [256]
    float* shift  = scale + HID;                    // [256]

    // ---- degrees (once, shared by all 3 layers) ----
    zero_kernel<<<512, 256, 0, stream>>>(dinv, N_NODES);
    zero_kernel<<<128, 256, 0, stream>>>(binv, N_HE);
    degree_kernel<<<(N_INC + 255) / 256, 256, 0, stream>>>(node_idx, he_idx, dinv, binv);
    invert_kernel<<<(N_NODES + 255) / 256, 256, 0, stream>>>(dinv, N_NODES);
    invert_kernel<<<(N_HE + 255) / 256, 256, 0, stream>>>(binv, N_HE);

    // ---- layer 1: x[.,128] @ W1 -> 256, hconv, relu, BN ----
    {
        const int F = HID, gshift = 6;
        dim3 gg(N_NODES / 16, F / 64);
        gemm_wmma<<<gg, 32, 0, stream>>>(x, W1, xw, IN_CH, F);
        zero_kernel<<<2048, 256, 0, stream>>>(heb, N_HE * F);
        int total = N_INC << gshift;
        scatter_fwd<<<(total + 255) / 256, 256, 0, stream>>>(xw, node_idx, he_idx, heb, F, gshift);
        zero_kernel<<<4096, 256, 0, stream>>>(h, N_NODES * F);
        scatter_bwd<<<(total + 255) / 256, 256, 0, stream>>>(heb, binv, node_idx, he_idx, h, F, gshift);
        zero_kernel<<<2, 256, 0, stream>>>(csum, 2 * HID);  // csum + csumsq contiguous
        epilogue_bn<<<N_NODES / BN_CHUNK, 256, 0, stream>>>(h, dinv, b1, csum, csumsq);
        bn_stats<<<1, 256, 0, stream>>>(csum, csumsq, g1, be1, scale, shift);
        bn_apply<<<N_NODES, 256, 0, stream>>>(h, scale, shift);
    }

    // ---- layer 2: h[.,256] @ W2 -> 256, hconv, relu, BN ----
    {
        const int F = HID, gshift = 6;
        dim3 gg(N_NODES / 16, F / 64);
        gemm_wmma<<<gg, 32, 0, stream>>>(h, W2, xw, HID, F);
        zero_kernel<<<2048, 256, 0, stream>>>(heb, N_HE * F);
        int total = N_INC << gshift;
        scatter_fwd<<<(total + 255) / 256, 256, 0, stream>>>(xw, node_idx, he_idx, heb, F, gshift);
        zero_kernel<<<4096, 256, 0, stream>>>(h, N_NODES * F);
        scatter_bwd<<<(total + 255) / 256, 256, 0, stream>>>(heb, binv, node_idx, he_idx, h, F, gshift);
        zero_kernel<<<2, 256, 0, stream>>>(csum, 2 * HID);
        epilogue_bn<<<N_NODES / BN_CHUNK, 256, 0, stream>>>(h, dinv, b2, csum, csumsq);
        bn_stats<<<1, 256, 0, stream>>>(csum, csumsq, g2, be2, scale, shift);
        bn_apply<<<N_NODES, 256, 0, stream>>>(h, scale, shift);
    }

    // ---- layer 3: h[.,256] @ W3 -> 128, hconv, relu -> out ----
    {
        const int F = OUT_CH, gshift = 5;
        dim3 gg(N_NODES / 16, F / 64);
        gemm_wmma<<<gg, 32, 0, stream>>>(h, W3, xw, HID, F);
        zero_kernel<<<1024, 256, 0, stream>>>(heb, N_HE * F);
        int total = N_INC << gshift;
        scatter_fwd<<<(total + 255) / 256, 256, 0, stream>>>(xw, node_idx, he_idx, heb, F, gshift);
        zero_kernel<<<2048, 256, 0, stream>>>(h, N_NODES * F);
        scatter_bwd<<<(total + 255) / 256, 256, 0, stream>>>(heb, binv, node_idx, he_idx, h, F, gshift);
        epilogue_out<<<N_NODES, 128, 0, stream>>>(h, dinv, b3, out);
    }
}